// AttnHGCN_22136261444077
// MI455X (gfx1250) — compile-verified
//
#include <hip/hip_runtime.h>
#include <stdint.h>

// ---------------------------------------------------------------------------
// AttnHGCN.forward_ui on MI455X (gfx1250).
// Memory-bound scatter/gather: embeddings (38.4 MB) + edge stream (24 MB) are
// L2-resident (192 MB), so per-edge gathers hit L2 and scatter-adds resolve in
// L2 atomic units via hardware global_atomic_add_f32 (scope DEV). Edge tuples
// are staged per 256-edge tile into LDS with CDNA5 async-to-LDS b128 bursts
// (ASYNCcnt). Wave32 layout: 1 wave per edge, 1 float2 per lane covers the
// 64-channel row in one 256B access.
// ---------------------------------------------------------------------------

#define CHANNEL 64

__device__ __forceinline__ void gatomic_add2_f32(float* p, float x, float y) {
    // Two channel atomics sharing one 64-bit address: base and base+4.
    asm volatile("global_atomic_add_f32 %0, %1, off scope:SCOPE_DEV\n\t"
                 "global_atomic_add_f32 %0, %2, off offset:4 scope:SCOPE_DEV"
                 :: "v"(p), "v"(x), "v"(y) : "memory");
}

__device__ __forceinline__ void async_load_b32_to_lds(unsigned lds_off, const void* gptr) {
    asm volatile("global_load_async_to_lds_b32 %0, %1, off"
                 :: "v"(lds_off), "v"(gptr) : "memory");
}

__device__ __forceinline__ void async_load_b128_to_lds(unsigned lds_off, const void* gptr) {
    asm volatile("global_load_async_to_lds_b128 %0, %1, off"
                 :: "v"(lds_off), "v"(gptr) : "memory");
}

__device__ __forceinline__ void wait_asynccnt0() {
    asm volatile("s_wait_asynccnt 0" ::: "memory");
}

// Stage a 4-element (16B) chunk of one edge array into LDS.
__device__ __forceinline__ void stage4(void* lds_ptr, const void* gptr, int c, int n_edges) {
    if (c + 3 < n_edges) {
        async_load_b128_to_lds((unsigned)(uintptr_t)lds_ptr, gptr);
    } else {
        #pragma unroll
        for (int m = 0; m < 4; ++m)
            if (c + m < n_edges)
                async_load_b32_to_lds((unsigned)(uintptr_t)lds_ptr + 4u * m,
                                      (const char*)gptr + 4 * m);
    }
}

// out = input embeddings (residual term 0); zero the layer-0 scatter targets.
__global__ __launch_bounds__(256)
void hgcn_init_kernel(const float4* __restrict__ user_emb,
                      const float4* __restrict__ item_emb,
                      float4* __restrict__ out_user, float4* __restrict__ out_item,
                      float4* __restrict__ u_next, float4* __restrict__ it_next,
                      int n_user_vec4, int n_item_vec4) {
    int i = blockIdx.x * blockDim.x + threadIdx.x;
    float4 z; z.x = 0.f; z.y = 0.f; z.z = 0.f; z.w = 0.f;
    if (i < n_user_vec4) { out_user[i] = user_emb[i]; u_next[i] = z; }
    if (i < n_item_vec4) { out_item[i] = item_emb[i]; it_next[i] = z; }
}

// One wave (32 lanes) per edge; lane l covers channels [2l, 2l+1] as float2.
// 256 edges staged per block through LDS via async b128 loads.
__global__ __launch_bounds__(256)
void hgcn_edge_kernel(const float* __restrict__ u_cur,
                      const float* __restrict__ it_cur,
                      float* __restrict__ u_next, float* __restrict__ it_next,
                      const int* __restrict__ u_idx, const int* __restrict__ i_idx,
                      const float* __restrict__ w, int n_edges) {
    __shared__ __align__(16) int   su[256];
    __shared__ __align__(16) int   si[256];
    __shared__ __align__(16) float sw[256];

    const int t    = threadIdx.x;
    const int base = blockIdx.x * 256;

    // 192 threads each stage one 16B chunk (4 edges) of one array.
    if (t < 64) {
        const int j = (t & 63) * 4, c = base + j;
        if (c < n_edges) stage4(&su[j], u_idx + c, c, n_edges);
    } else if (t < 128) {
        const int j = (t & 63) * 4, c = base + j;
        if (c < n_edges) stage4(&si[j], i_idx + c, c, n_edges);
    } else if (t < 192) {
        const int j = (t & 63) * 4, c = base + j;
        if (c < n_edges) stage4(&sw[j], w + c, c, n_edges);
    }
    wait_asynccnt0();
    __syncthreads();

    const int wave = t >> 5;
    const int lane = t & 31;
    int nloc = n_edges - base;
    if (nloc > 256) nloc = 256;

    #pragma unroll 2
    for (int k = 0; k < 32; ++k) {
        const int idx = wave * 32 + k;          // wave-uniform
        if (idx >= nloc) break;                 // wave-uniform exit

        // Scalarize edge tuple -> SGPRs so gathers use saddr-form addressing.
        const int   uu = __builtin_amdgcn_readfirstlane(su[idx]);
        const int   ii = __builtin_amdgcn_readfirstlane(si[idx]);
        const float ww = __int_as_float(
            __builtin_amdgcn_readfirstlane(__float_as_int(sw[idx])));

        const float2 iv = ((const float2*)(it_cur + (size_t)ii * CHANNEL))[lane];
        const float2 uv = ((const float2*)(u_cur  + (size_t)uu * CHANNEL))[lane];

        float* up = u_next  + (size_t)uu * CHANNEL + 2 * lane;
        float* ip = it_next + (size_t)ii * CHANNEL + 2 * lane;

        gatomic_add2_f32(up, ww * iv.x, ww * iv.y);   // users += w * item[i_idx]
        gatomic_add2_f32(ip, ww * uv.x, ww * uv.y);   // items += w * user[u_idx]
    }
}

// Residual accumulate; optionally zero the buffer that serves as "next"
// for the next layer; final layer folds the 1/(L+1) scale.
__global__ __launch_bounds__(256)
void hgcn_accum_kernel(float4* __restrict__ out_user, float4* __restrict__ out_item,
                       const float4* __restrict__ u_next, const float4* __restrict__ it_next,
                       float4* __restrict__ u_zero, float4* __restrict__ it_zero,
                       float scale, int n_user_vec4, int n_item_vec4) {
    int i = blockIdx.x * blockDim.x + threadIdx.x;
    float4 z; z.x = 0.f; z.y = 0.f; z.z = 0.f; z.w = 0.f;
    if (i < n_user_vec4) {
        float4 o = out_user[i], a = u_next[i];
        o.x = (o.x + a.x) * scale; o.y = (o.y + a.y) * scale;
        o.z = (o.z + a.z) * scale; o.w = (o.w + a.w) * scale;
        out_user[i] = o;
        if (u_zero) u_zero[i] = z;
    }
    if (i < n_item_vec4) {
        float4 o = out_item[i], a = it_next[i];
        o.x = (o.x + a.x) * scale; o.y = (o.y + a.y) * scale;
        o.z = (o.z + a.z) * scale; o.w = (o.w + a.w) * scale;
        out_item[i] = o;
        if (it_zero) it_zero[i] = z;
    }
}

extern "C" void kernel_launch(void* const* d_in, const int* in_sizes, int n_in,
                              void* d_out, int out_size, void* d_ws, size_t ws_size,
                              hipStream_t stream) {
    (void)n_in; (void)out_size; (void)ws_size;

    // Inputs: [0]=layers_num(1), [1]=user_emb, [2]=item_emb,
    //         [3]=inter_edge(2,E) int32, [4]=inter_edge_w(E)
    const float* user_emb  = (const float*)d_in[1];
    const float* item_emb  = (const float*)d_in[2];
    const int*   inter     = (const int*)d_in[3];
    const float* edge_w    = (const float*)d_in[4];

    const int n_user_elems = in_sizes[1];        // N_USERS * 64
    const int n_item_elems = in_sizes[2];        // N_ITEMS * 64
    const int n_edges      = in_sizes[4];
    const int* u_idx = inter;
    const int* i_idx = inter + n_edges;

    const int layers = 3;                        // reference LAYERS constant
    const float inv  = 1.0f / (float)(layers + 1);

    // Output layout matches the reference tuple: (item_acc, user_acc).
    float* out_item = (float*)d_out;
    float* out_user = out_item + n_item_elems;

    // Workspace: double-buffered layer state (2x users + 2x items ~ 76.8 MB).
    char* ws = (char*)d_ws;
    float* ubuf[2];
    float* ibuf[2];
    ubuf[0] = (float*)ws;                        ws += (size_t)n_user_elems * sizeof(float);
    ubuf[1] = (float*)ws;                        ws += (size_t)n_user_elems * sizeof(float);
    ibuf[0] = (float*)ws;                        ws += (size_t)n_item_elems * sizeof(float);
    ibuf[1] = (float*)ws;

    const int n_user_vec4 = n_user_elems / 4;    // CHANNEL=64 => divisible by 4
    const int n_item_vec4 = n_item_elems / 4;

    const int blk   = 256;
    const int gElem = (n_user_vec4 + blk - 1) / blk;     // covers items too
    const int gEdge = (n_edges + 255) / 256;

    hgcn_init_kernel<<<gElem, blk, 0, stream>>>((const float4*)user_emb,
                                                (const float4*)item_emb,
                                                (float4*)out_user, (float4*)out_item,
                                                (float4*)ubuf[0], (float4*)ibuf[0],
                                                n_user_vec4, n_item_vec4);

    for (int l = 0; l < layers; ++l) {
        const bool last = (l == layers - 1);
        // layer l: cur = inputs (l==0) or previous next; next ping-pongs a/b.
        const float* uc = (l == 0) ? user_emb : ubuf[(l + 1) & 1];
        const float* ic = (l == 0) ? item_emb : ibuf[(l + 1) & 1];
        float* un  = ubuf[l & 1];
        float* in_ = ibuf[l & 1];
        float* uz  = last ? nullptr : ubuf[(l + 1) & 1]; // next layer's scatter target
        float* iz  = last ? nullptr : ibuf[(l + 1) & 1];

        hgcn_edge_kernel<<<gEdge, 256, 0, stream>>>(uc, ic, un, in_,
                                                    u_idx, i_idx, edge_w, n_edges);

        hgcn_accum_kernel<<<gElem, blk, 0, stream>>>((float4*)out_user, (float4*)out_item,
                                                     (const float4*)un, (const float4*)in_,
                                                     (float4*)uz, (float4*)iz,
                                                     last ? inv : 1.0f,
                                                     n_user_vec4, n_item_vec4);
    }
}